// _Constrained_Affine_All_5188320493998
// MI455X (gfx1250) — compile-verified
//
#include <hip/hip_runtime.h>
#include <math.h>

typedef float v4f __attribute__((ext_vector_type(4)));
typedef float v2f __attribute__((ext_vector_type(2)));
typedef __attribute__((aligned(4))) v2f v2f_u;   // 4B-aligned b64 load (legal in DWORD align mode)

#define Wd   224
#define Hd   224
#define HWp  (Hd * Wd)         // 50176
#define CHW  (3 * HWp)         // 150528
#define Nn   8
#define KAUG 48                // 3 * 16 samples
#define NPLANE (Nn * KAUG)     // 384
#define BPP  ((HWp / 4) / 256) // 49 blocks per plane

__device__ __forceinline__ int iclamp(int v, int lo, int hi) {
    return v < lo ? lo : (v > hi ? hi : v);
}

// ---------------------------------------------------------------------------
// Setup: fold affine_grid (both align_corners modes) + grid_sample
// un-normalization into 6 linear coefficients per augmentation plane.
//   plane g = n*48 + k  (reference's concatenated grid index)
//   g < 128: rotation (ac=False) | g < 256: translation (ac=True) | else scale
// ---------------------------------------------------------------------------
__global__ void setup_coeffs(const float* __restrict__ u_theta,
                             const float* __restrict__ u_r1,
                             const float* __restrict__ u_r2,
                             const float* __restrict__ u_s1,
                             float* __restrict__ coef) {
    int g = blockIdx.x * blockDim.x + threadIdx.x;
    if (g >= NPLANE) return;

    float M00, M01, M02, M10, M11, M12;
    float fx = 1.0f, fy = 1.0f;
    if (g < 128) {
        float th = (u_theta[g] * 2.0f - 1.0f) * 3.14159265358979323846f;
        float c = cosf(th), s = sinf(th);
        M00 = c;  M01 = -s; M02 = 0.0f;
        M10 = s;  M11 = c;  M12 = 0.0f;
        fx = (float)(Wd - 1) / (float)Wd;
        fy = (float)(Hd - 1) / (float)Hd;
    } else if (g < 256) {
        int j = g - 128;
        M00 = 1.0f; M01 = 0.0f; M02 = u_r1[j] * 2.0f - 1.0f;
        M10 = 0.0f; M11 = 1.0f; M12 = u_r2[j] * 2.0f - 1.0f;
    } else {
        int j = g - 256;
        float s1 = expf(u_s1[j] * 2.0f - 1.0f);
        M00 = s1;   M01 = 0.0f; M02 = 0.0f;
        M10 = 0.0f; M11 = s1;   M12 = 0.0f;
    }

    float P = M00 * fx, Q = M01 * fy, R = M02;
    float S = M10 * fx, T = M11 * fy, U = M12;

    float a1 = (float)Wd * P / (float)(Wd - 1);
    float a2 = (float)Wd * Q / (float)(Hd - 1);
    float a0 = 0.5f * (float)Wd * (R - P - Q) + 0.5f * (float)(Wd - 1);
    float b1 = (float)Hd * S / (float)(Wd - 1);
    float b2 = (float)Hd * T / (float)(Hd - 1);
    float b0 = 0.5f * (float)Hd * (U - S - T) + 0.5f * (float)(Hd - 1);

    float* o = coef + g * 8;
    o[0] = a0; o[1] = a1; o[2] = a2;
    o[3] = b0; o[4] = b1; o[5] = b2;
    o[6] = 0.0f; o[7] = 0.0f;
}

// ---------------------------------------------------------------------------
// Main warp kernel: each thread -> 4 consecutive pixels x 3 channels.
// Paired-corner b64 gathers (x-corners are adjacent) + cndmask lane selects;
// NT b128 stores keep the 236 MB output stream out of L2/WGP$ so the 4.8 MB
// of input images stays cache-resident across all 48 warps per image.
// ---------------------------------------------------------------------------
__global__ void __launch_bounds__(256) warp_sample_kernel(
        const float* __restrict__ xin,
        const float* __restrict__ coef,
        float* __restrict__ out) {
    int bid   = blockIdx.x;
    int plane = bid / BPP;            // 0..383 (== g == n*48 + k)
    int pb    = bid % BPP;
    int n     = plane / KAUG;
    int k     = plane % KAUG;

    const float* cf = coef + plane * 8;   // block-uniform -> scalar loads
    float a0 = cf[0], a1 = cf[1], a2 = cf[2];
    float b0 = cf[3], b1 = cf[4], b2 = cf[5];

    int q  = pb * 256 + (int)threadIdx.x;
    int p0 = q * 4;                        // W%4==0 -> 4 pixels share a row
    int h  = p0 / Wd;
    int w0 = p0 % Wd;

    const float* img = xin + (size_t)n * CHW;

    float ixb = a0 + a1 * (float)w0 + a2 * (float)h;
    float iyb = b0 + b1 * (float)w0 + b2 * (float)h;

    {   // prefetch hint near this thread's first sample point
        int py = iclamp((int)iyb, 0, Hd - 1);
        int px = iclamp((int)ixb, 0, Wd - 1);
        __builtin_prefetch(img + py * Wd + px, 0, 0);
    }

    float fo0[4], fo1[4], fo2[4];
#pragma unroll
    for (int i = 0; i < 4; ++i) {
        float ix = ixb + a1 * (float)i;
        float iy = iyb + b1 * (float)i;
        float x0f = floorf(ix), y0f = floorf(iy);
        float wx1 = ix - x0f, wx0 = 1.0f - wx1;
        float wy1 = iy - y0f, wy0 = 1.0f - wy1;
        int x0 = (int)x0f, y0 = (int)y0f;
        int x1 = x0 + 1,   y1 = y0 + 1;

        float vx0 = (x0 >= 0 && x0 < Wd) ? 1.0f : 0.0f;
        float vx1 = (x1 >= 0 && x1 < Wd) ? 1.0f : 0.0f;
        float vy0 = (y0 >= 0 && y0 < Hd) ? 1.0f : 0.0f;
        float vy1 = (y1 >= 0 && y1 < Hd) ? 1.0f : 0.0f;

        float w00 = wx0 * wy0 * vx0 * vy0;
        float w10 = wx1 * wy0 * vx1 * vy0;
        float w01 = wx0 * wy1 * vx0 * vy1;
        float w11 = wx1 * wy1 * vx1 * vy1;

        int y0c = iclamp(y0, 0, Hd - 1), y1c = iclamp(y1, 0, Hd - 1);
        // Paired corners: load {xl, xl+1}; any lane whose true corner is
        // outside [xl, xl+1] carries weight 0, so the select fallback is safe.
        int  xl  = iclamp(x0, 0, Wd - 2);
        bool c00 = (x0 == xl);        // v00 = lo else hi
        bool c10 = (x1 == xl + 1);    // v10 = hi else lo
        int  o0  = y0c * Wd + xl;
        int  o1  = y1c * Wd + xl;

        const float* p0c = img;
        const float* p1c = img + HWp;
        const float* p2c = img + 2 * HWp;

        v2f t0 = *(const v2f_u*)(p0c + o0); v2f b0v = *(const v2f_u*)(p0c + o1);
        v2f t1 = *(const v2f_u*)(p1c + o0); v2f b1v = *(const v2f_u*)(p1c + o1);
        v2f t2 = *(const v2f_u*)(p2c + o0); v2f b2v = *(const v2f_u*)(p2c + o1);

        fo0[i] = w00 * (c00 ? t0.x : t0.y) + w10 * (c10 ? t0.y : t0.x)
               + w01 * (c00 ? b0v.x : b0v.y) + w11 * (c10 ? b0v.y : b0v.x);
        fo1[i] = w00 * (c00 ? t1.x : t1.y) + w10 * (c10 ? t1.y : t1.x)
               + w01 * (c00 ? b1v.x : b1v.y) + w11 * (c10 ? b1v.y : b1v.x);
        fo2[i] = w00 * (c00 ? t2.x : t2.y) + w10 * (c10 ? t2.y : t2.x)
               + w01 * (c00 ? b2v.x : b2v.y) + w11 * (c10 ? b2v.y : b2v.x);
    }

    float* ob = out + ((size_t)(n * 49 + (k + 1))) * CHW + p0;
    v4f v0 = { fo0[0], fo0[1], fo0[2], fo0[3] };
    v4f v1 = { fo1[0], fo1[1], fo1[2], fo1[3] };
    v4f v2 = { fo2[0], fo2[1], fo2[2], fo2[3] };
    __builtin_nontemporal_store(v0, (v4f*)(ob));
    __builtin_nontemporal_store(v1, (v4f*)(ob + HWp));
    __builtin_nontemporal_store(v2, (v4f*)(ob + 2 * HWp));
}

// ---------------------------------------------------------------------------
// Copy original images into slot k = 0 using the CDNA5 async-LDS pipeline:
// global_load_async_to_lds_b128 -> s_wait_asynccnt -> store_async_from_lds.
// Grid is exactly 1176 blocks * 256 threads = 301056 float4s (no tail).
// ---------------------------------------------------------------------------
__device__ __forceinline__ unsigned lds_byte_offset(const void* p) {
    return (unsigned)(unsigned long long)
        (__attribute__((address_space(3))) const char*)(const char*)p;
}

__global__ void __launch_bounds__(256) copy_orig_kernel(
        const float* __restrict__ xin, float* __restrict__ out) {
    __shared__ char lbuf[256 * 16];
    int i = blockIdx.x * 256 + (int)threadIdx.x;   // float4 index, exact fit
    int e = i * 4;
    int n = e / CHW;
    int r = e - n * CHW;

    unsigned lds_off = lds_byte_offset(&lbuf[threadIdx.x * 16]);
    unsigned long long gsrc = (unsigned long long)(xin + e);
    unsigned long long gdst = (unsigned long long)(out + (size_t)n * 49 * CHW + r);

    asm volatile("global_load_async_to_lds_b128 %0, %1, off"
                 :: "v"(lds_off), "v"(gsrc) : "memory");
    asm volatile("s_wait_asynccnt 0x0" ::: "memory");
    asm volatile("global_store_async_from_lds_b128 %0, %1, off"
                 :: "v"(gdst), "v"(lds_off) : "memory");
    asm volatile("s_wait_asynccnt 0x0" ::: "memory");
}

extern "C" void kernel_launch(void* const* d_in, const int* in_sizes, int n_in,
                              void* d_out, int out_size, void* d_ws, size_t ws_size,
                              hipStream_t stream) {
    const float* x       = (const float*)d_in[0];
    const float* u_theta = (const float*)d_in[1];
    const float* u_r1    = (const float*)d_in[2];
    const float* u_r2    = (const float*)d_in[3];
    const float* u_s1    = (const float*)d_in[4];
    float* out  = (float*)d_out;
    float* coef = (float*)d_ws;   // 384 * 8 floats = 12 KB

    setup_coeffs<<<(NPLANE + 127) / 128, 128, 0, stream>>>(u_theta, u_r1, u_r2, u_s1, coef);

    int copy_blocks = (Nn * CHW / 4) / 256;        // 1176 exact
    copy_orig_kernel<<<copy_blocks, 256, 0, stream>>>(x, out);

    int main_blocks = NPLANE * BPP;                // 384 * 49 = 18816
    warp_sample_kernel<<<main_blocks, 256, 0, stream>>>(x, coef, out);
}